// ActorCritic_53171695125335
// MI455X (gfx1250) — compile-verified
//
#include <hip/hip_runtime.h>

// Problem constants (from reference)
#define T_    512
#define B_    256
#define IN_   292
#define H_    128
#define A_    12
#define G4    512          // 4*H
#define OUTW  13           // A+1
#define KIH   73           // IN/4  (K-steps of 4 for x part)
#define KHH   32           // H/4   (K-steps of 4 for h part)
#define KC    8            // k-steps per LDS chunk in pre_gemm (32 floats)
#define NCHUNK 9           // 9*8 = 72 ksteps staged, + 1 remainder kstep
#define LDP   36           // LDS row stride: 144B = 16B aligned, conflict-free

typedef __attribute__((ext_vector_type(2))) float v2f;
typedef __attribute__((ext_vector_type(4))) float v4f;
typedef __attribute__((ext_vector_type(8))) float v8f;
typedef __attribute__((ext_vector_type(4))) unsigned int u32x4;
typedef __attribute__((ext_vector_type(4))) int i32x4;
typedef __attribute__((ext_vector_type(8))) int i32x8;

#if defined(__has_builtin)
#if __has_builtin(__builtin_amdgcn_tensor_load_to_lds) && \
    __has_builtin(__builtin_amdgcn_s_wait_tensorcnt)
#define HAVE_TDM 1
#endif
#endif

// ---- CDNA5 fp32 WMMA: D = A(16x4) * B(4x16) + C(16x16) ----
static __device__ __forceinline__ v8f wmma4(v2f a, v2f b, v8f c) {
  return __builtin_amdgcn_wmma_f32_16x16x4_f32(false, a, false, b, (short)0, c,
                                               false, false);
}

static __device__ __forceinline__ v8f splat8(float s) {
  v8f r;
#pragma unroll
  for (int v = 0; v < 8; ++v) r[v] = s;
  return r;
}

static __device__ __forceinline__ float sigm(float x) {
  return 1.0f / (1.0f + __expf(-x));
}
static __device__ __forceinline__ float tanh_(float x) {
  float e = __expf(2.0f * x);
  return (e - 1.0f) / (e + 1.0f);
}

#if defined(HAVE_TDM)
// Issue one TDM 2D tile load: tile (tile_d1 rows x 32 floats) from a
// row-major fp32 tensor (row stride 292 floats) into LDS, hardware-padding
// each 32-dword row by 4 dwords -> LDS row stride 36 floats (LDP).
// D# bitfields per CDNA5 ISA ch.8 (group0: count/lds/global/type,
// group1: data_size, pad_enable, pad_interval=32dw, pad_amount=4dw, dims).
static __device__ __forceinline__ void tdm_load_2d(unsigned lds_byte_off,
                                                   const void* gaddr,
                                                   unsigned tensor_d1,
                                                   unsigned tile_d1) {
  unsigned long long ga = (unsigned long long)(size_t)gaddr;
  u32x4 g0 = {};
  g0[0] = 1u;                                                  // count=1
  g0[1] = lds_byte_off;                                        // lds_addr
  g0[2] = (unsigned)(ga & 0xFFFFFFFFu);                        // global lo
  g0[3] = ((unsigned)(ga >> 32) & 0x01FFFFFFu) | (2u << 30);   // hi25 | type=2
  i32x8 g1 = {};
  g1[0] = (int)((2u << 16) | (1u << 20) | (4u << 22) | (3u << 25));
  g1[1] = (int)(292u << 16);                       // tensor_dim0 = 292
  g1[2] = (int)((tensor_d1 & 0xFFFFu) << 16);      // tensor_dim1 lo16
  g1[3] = (int)(((tensor_d1 >> 16) & 0xFFFFu) | (32u << 16));  // tile_dim0=32
  g1[4] = (int)tile_d1;                            // tile_dim1 (tile_dim2=0)
  g1[5] = 292;                                     // tensor_dim0_stride
  g1[6] = 0;
  g1[7] = 0;
  i32x4 z4 = {};
  i32x8 z8 = {};
#if __clang_major__ >= 23
  __builtin_amdgcn_tensor_load_to_lds(g0, g1, z4, z4, z8, 0);
#else
  (void)z8;
  __builtin_amdgcn_tensor_load_to_lds(g0, g1, z4, z4, 0);
#endif
}
#endif

// ============================================================
// Kernel 1: gates_pre = x @ W_ih^T + (b_ih + b_hh).
// Block = 64 rows x 128 gate-cols; K in 32-float chunks, DOUBLE-BUFFERED:
// chunk ch+1 streams into LDS (TDM DMA if available, else pipelined b128
// loads) while chunk ch feeds the WMMAs. One barrier per chunk.
// Wave = 2 M-tiles x 2 N-tiles. Output in D-fragment tile order:
//   gpre[(mt*32+nt)*256 + vgpr*32 + lane]  -> coalesced 128B lines.
// ============================================================
__global__ __launch_bounds__(256) void pre_gemm(
    const float* __restrict__ x, const float* __restrict__ Wih,
    const float* __restrict__ bih, const float* __restrict__ bhh,
    float* __restrict__ gpre) {
  __shared__ float sx[2][64][LDP];    // 18 KB
  __shared__ float sw[2][128][LDP];   // 36 KB

  const int tid  = threadIdx.x;
  const int lane = tid & 31;
  const int wid  = tid >> 5;
  const int lm   = lane & 15;
  const int lh   = lane >> 4;
  const int koff = 2 * lh;

  const int mb   = blockIdx.x >> 2;  // 2048 row-blocks of 64
  const int nb   = blockIdx.x & 3;   // 4 col-blocks of 128
  const int row0 = mb * 64;
  const int col0 = nb * 128;

  const int mg = wid & 1;   // M-tiles {2mg, 2mg+1}
  const int ng = wid >> 1;  // N-tiles {2ng, 2ng+1}

  v8f acc[2][2];
#pragma unroll
  for (int i = 0; i < 2; ++i)
#pragma unroll
    for (int j = 0; j < 2; ++j) {
      int n = col0 + (2 * ng + j) * 16 + lm;
      acc[i][j] = splat8(bih[n] + bhh[n]);
    }

#if defined(HAVE_TDM)
  const unsigned sx_base = (unsigned)(size_t)&sx[0][0][0];
  const unsigned sw_base = (unsigned)(size_t)&sw[0][0][0];
  const unsigned SXB = 64u * LDP * 4u, SWB = 128u * LDP * 4u;
  if (wid == 0) {  // preload chunk 0 via Tensor Data Mover
    tdm_load_2d(sx_base, x + (size_t)row0 * IN_, (unsigned)(T_ * B_), 64u);
    tdm_load_2d(sw_base, Wih + (size_t)col0 * IN_, (unsigned)G4, 128u);
  }
  for (int ch = 0; ch < NCHUNK; ++ch) {
    if (wid == 0) __builtin_amdgcn_s_wait_tensorcnt(0);
    __syncthreads();  // chunk ch resident; chunk ch-1 fully consumed
    if (wid == 0 && ch + 1 < NCHUNK) {
      tdm_load_2d(sx_base + ((ch + 1) & 1) * SXB,
                  x + (size_t)row0 * IN_ + (ch + 1) * 32, (unsigned)(T_ * B_),
                  64u);
      tdm_load_2d(sw_base + ((ch + 1) & 1) * SWB,
                  Wih + (size_t)col0 * IN_ + (ch + 1) * 32, (unsigned)G4,
                  128u);
    }
    const int buf = ch & 1;
#pragma unroll
    for (int kk = 0; kk < KC; ++kk) {
      v2f a0 = *(const v2f*)(&sx[buf][(2 * mg + 0) * 16 + lm][4 * kk + koff]);
      v2f a1 = *(const v2f*)(&sx[buf][(2 * mg + 1) * 16 + lm][4 * kk + koff]);
      v2f b0 = *(const v2f*)(&sw[buf][(2 * ng + 0) * 16 + lm][4 * kk + koff]);
      v2f b1 = *(const v2f*)(&sw[buf][(2 * ng + 1) * 16 + lm][4 * kk + koff]);
      acc[0][0] = wmma4(a0, b0, acc[0][0]);
      acc[0][1] = wmma4(a0, b1, acc[0][1]);
      acc[1][0] = wmma4(a1, b0, acc[1][0]);
      acc[1][1] = wmma4(a1, b1, acc[1][1]);
    }
  }
#else
  // Manual pipeline: b128 loads into registers one chunk ahead.
  v4f rx[2], rw[4];
#pragma unroll
  for (int i = 0; i < 2; ++i) {
    int idx = tid + i * 256, r = idx >> 3, c4 = idx & 7;
    rx[i] = *(const v4f*)(x + (size_t)(row0 + r) * IN_ + c4 * 4);
  }
#pragma unroll
  for (int i = 0; i < 4; ++i) {
    int idx = tid + i * 256, r = idx >> 3, c4 = idx & 7;
    rw[i] = *(const v4f*)(Wih + (size_t)(col0 + r) * IN_ + c4 * 4);
  }
  for (int ch = 0; ch < NCHUNK; ++ch) {
    const int buf = ch & 1;
#pragma unroll
    for (int i = 0; i < 2; ++i) {
      int idx = tid + i * 256, r = idx >> 3, c4 = idx & 7;
      *(v4f*)(&sx[buf][r][c4 * 4]) = rx[i];
    }
#pragma unroll
    for (int i = 0; i < 4; ++i) {
      int idx = tid + i * 256, r = idx >> 3, c4 = idx & 7;
      *(v4f*)(&sw[buf][r][c4 * 4]) = rw[i];
    }
    __syncthreads();
    if (ch + 1 < NCHUNK) {
#pragma unroll
      for (int i = 0; i < 2; ++i) {
        int idx = tid + i * 256, r = idx >> 3, c4 = idx & 7;
        rx[i] = *(const v4f*)(x + (size_t)(row0 + r) * IN_ + (ch + 1) * 32 +
                              c4 * 4);
      }
#pragma unroll
      for (int i = 0; i < 4; ++i) {
        int idx = tid + i * 256, r = idx >> 3, c4 = idx & 7;
        rw[i] = *(const v4f*)(Wih + (size_t)(col0 + r) * IN_ + (ch + 1) * 32 +
                              c4 * 4);
      }
    }
#pragma unroll
    for (int kk = 0; kk < KC; ++kk) {
      v2f a0 = *(const v2f*)(&sx[buf][(2 * mg + 0) * 16 + lm][4 * kk + koff]);
      v2f a1 = *(const v2f*)(&sx[buf][(2 * mg + 1) * 16 + lm][4 * kk + koff]);
      v2f b0 = *(const v2f*)(&sw[buf][(2 * ng + 0) * 16 + lm][4 * kk + koff]);
      v2f b1 = *(const v2f*)(&sw[buf][(2 * ng + 1) * 16 + lm][4 * kk + koff]);
      acc[0][0] = wmma4(a0, b0, acc[0][0]);
      acc[0][1] = wmma4(a0, b1, acc[0][1]);
      acc[1][0] = wmma4(a1, b0, acc[1][0]);
      acc[1][1] = wmma4(a1, b1, acc[1][1]);
    }
  }
#endif
  {  // remainder k-step 72 (K = 288..291) straight from global
    const int kb = 288 + koff;
    v2f a0 = *(const v2f*)(x + (size_t)(row0 + (2 * mg + 0) * 16 + lm) * IN_ + kb);
    v2f a1 = *(const v2f*)(x + (size_t)(row0 + (2 * mg + 1) * 16 + lm) * IN_ + kb);
#pragma unroll
    for (int j = 0; j < 2; ++j) {
      v2f b = *(const v2f*)(Wih + (size_t)(col0 + (2 * ng + j) * 16 + lm) * IN_ + kb);
      acc[0][j] = wmma4(a0, b, acc[0][j]);
      acc[1][j] = wmma4(a1, b, acc[1][j]);
    }
  }
#pragma unroll
  for (int i = 0; i < 2; ++i)
#pragma unroll
    for (int j = 0; j < 2; ++j) {
      int mt = (row0 >> 4) + 2 * mg + i;
      int nt = (col0 >> 4) + 2 * ng + j;
      float* dst = gpre + ((size_t)mt * 32 + nt) * 256 + lane;
#pragma unroll
      for (int v = 0; v < 8; ++v) dst[v * 32] = acc[i][j][v];
    }
}

// ============================================================
// Kernel 2: T=512 recurrence + fused head. 16 WGs x 8 waves.
// Wave j owns hidden cols 16j..16j+15 (gate col-tiles j,j+8,j+16,j+24):
// i/f/g/o + c live in registers; W_hh fragments preloaded in VGPRs;
// done-mask applied per-lane to A-fragments (2 barriers per step).
// ============================================================
#define SHX(b, r, k) sh_x[((b) * 16 + (r)) * (IN_ + 8) + (k)]

template <int USE_PRE>
__global__ __launch_bounds__(256, 1) void lstm_seq(
    const float* __restrict__ x, const int* __restrict__ done,
    const float* __restrict__ h0, const float* __restrict__ c0,
    const float* __restrict__ Wih, const float* __restrict__ Whh,
    const float* __restrict__ bih, const float* __restrict__ bhh,
    const float* __restrict__ Wpi, const float* __restrict__ bpi,
    const float* __restrict__ Wv, const float* __restrict__ bvv,
    const float* __restrict__ gpre, float* __restrict__ out) {
  __shared__ float sh_h[16][H_ + 4];     // stride 132: conflict-free frags
  __shared__ float sh_head[16][H_ + 4];  // rows 0..11 W_pi, 12 W_v, 13..15 = 0
  __shared__ float sh_m[2][16];          // double-buffered reset mask
  __shared__ float sh_x[USE_PRE ? 1 : 2 * 16 * (IN_ + 8)];

  const int tid  = threadIdx.x;
  const int lane = tid & 31;
  const int wid  = tid >> 5;  // 0..7
  const int wg   = blockIdx.x;
  const int b0   = wg * 16;
  const int lm   = lane & 15;
  const int lh   = lane >> 4;
  const int koff = 2 * lh;
  const int hc   = wid * 16 + lm;  // hidden column owned by this lane

  for (int idx = tid; idx < 16 * H_; idx += 256) {
    int r = idx >> 7, k = idx & 127;
    sh_h[r][k] = h0[(size_t)(b0 + r) * H_ + k];
    float w = 0.f;
    if (r < A_) w = Wpi[r * H_ + k];
    else if (r == A_) w = Wv[k];
    sh_head[r][k] = w;
  }
  if (tid < 16) sh_m[0][tid] = 1.0f - (float)done[b0 + tid];
  if (!USE_PRE) {
    for (int idx = tid; idx < 16 * IN_; idx += 256) {
      int r = idx / IN_, k = idx - r * IN_;
      SHX(0, r, k) = x[(size_t)(b0 + r) * IN_ + k];
    }
  }

  v8f c;  // c-state tile: rows 8*lh+v, col hc
#pragma unroll
  for (int v = 0; v < 8; ++v) c[v] = c0[(size_t)(b0 + 8 * lh + v) * H_ + hc];

  const float bvi = bih[hc] + bhh[hc];
  const float bvf = bih[H_ + hc] + bhh[H_ + hc];
  const float bvg = bih[2 * H_ + hc] + bhh[2 * H_ + hc];
  const float bvo = bih[3 * H_ + hc] + bhh[3 * H_ + hc];
  const float bvh = (lm < A_) ? bpi[lm] : ((lm == A_) ? bvv[0] : 0.f);

  // preload ALL W_hh B-fragments into VGPRs (4 x 32 x v2f = 256 VGPRs)
  v2f whf[4][KHH];
#pragma unroll
  for (int g = 0; g < 4; ++g) {
    const float* wrow = Whh + (size_t)(g * H_ + hc) * H_ + koff;
#pragma unroll
    for (int kk = 0; kk < KHH; ++kk) whf[g][kk] = *(const v2f*)(wrow + kk * 4);
  }

  const float* wih_i = Wih + (size_t)(hc) * IN_ + koff;
  const float* wih_f = Wih + (size_t)(H_ + hc) * IN_ + koff;
  const float* wih_g = Wih + (size_t)(2 * H_ + hc) * IN_ + koff;
  const float* wih_o = Wih + (size_t)(3 * H_ + hc) * IN_ + koff;

  __syncthreads();

#pragma unroll 1
  for (int t = 0; t < T_; ++t) {
    const float mA = sh_m[t & 1][lm];  // reset mask for A-frag rows

    v8f ai, af, ag, ao;
    if (USE_PRE) {
      const float* gp = gpre + ((size_t)(t * 16 + wg) * 32) * 256 + lane;
#pragma unroll
      for (int v = 0; v < 8; ++v) {
        ai[v] = gp[(size_t)(wid) * 256 + v * 32];
        af[v] = gp[(size_t)(wid + 8) * 256 + v * 32];
        ag[v] = gp[(size_t)(wid + 16) * 256 + v * 32];
        ao[v] = gp[(size_t)(wid + 24) * 256 + v * 32];
      }
      if (t + 1 < T_) {  // prefetch next step's 4 tiles (one line per lane)
        const float* gn = gpre + ((size_t)((t + 1) * 16 + wg) * 32) * 256;
        int tj = wid + (lane >> 3) * 8;
        __builtin_prefetch(gn + (size_t)tj * 256 + (lane & 7) * 32, 0, 1);
      }
    } else {
      ai = splat8(bvi); af = splat8(bvf); ag = splat8(bvg); ao = splat8(bvo);
#pragma unroll 1
      for (int kk = 0; kk < KIH; ++kk) {
        v2f a = *(const v2f*)(&SHX(t & 1, lm, 4 * kk + koff));
        ai = wmma4(a, *(const v2f*)(wih_i + kk * 4), ai);
        af = wmma4(a, *(const v2f*)(wih_f + kk * 4), af);
        ag = wmma4(a, *(const v2f*)(wih_g + kk * 4), ag);
        ao = wmma4(a, *(const v2f*)(wih_o + kk * 4), ao);
      }
    }

#pragma unroll
    for (int v = 0; v < 8; ++v) c[v] *= sh_m[t & 1][8 * lh + v];

#pragma unroll
    for (int kk = 0; kk < KHH; ++kk) {
      v2f a = *(const v2f*)(&sh_h[lm][4 * kk + koff]);
      a = a * mA;
      ai = wmma4(a, whf[0][kk], ai);
      af = wmma4(a, whf[1][kk], af);
      ag = wmma4(a, whf[2][kk], ag);
      ao = wmma4(a, whf[3][kk], ao);
    }

    v8f hn;
#pragma unroll
    for (int v = 0; v < 8; ++v) {
      float cn = sigm(af[v]) * c[v] + sigm(ai[v]) * tanh_(ag[v]);
      c[v] = cn;
      hn[v] = sigm(ao[v]) * tanh_(cn);
    }

    if (tid < 16 && t + 1 < T_)
      sh_m[(t + 1) & 1][tid] = 1.0f - (float)done[(t + 1) * B_ + b0 + tid];
    if (!USE_PRE && t + 1 < T_) {
      for (int idx = tid; idx < 16 * IN_; idx += 256) {
        int r = idx / IN_, k = idx - r * IN_;
        SHX((t + 1) & 1, r, k) = x[((size_t)(t + 1) * B_ + b0 + r) * IN_ + k];
      }
    }

    __syncthreads();  // all waves done reading old sh_h
#pragma unroll
    for (int v = 0; v < 8; ++v) sh_h[8 * lh + v][hc] = hn[v];
    __syncthreads();  // new sh_h visible

    if (wid == (t & 7)) {  // rotating fused head: 16x16 tile, K=128
      v8f ha = splat8(bvh);
#pragma unroll
      for (int kk = 0; kk < KHH; ++kk) {
        v2f a = *(const v2f*)(&sh_h[lm][4 * kk + koff]);
        v2f b = *(const v2f*)(&sh_head[lm][4 * kk + koff]);
        ha = wmma4(a, b, ha);
      }
      if (lm < OUTW) {
#pragma unroll
        for (int v = 0; v < 8; ++v) {
          int r = 8 * lh + v;
          out[((size_t)t * B_ + b0 + r) * OUTW + lm] = ha[v];
        }
      }
    }
  }
}

extern "C" void kernel_launch(void* const* d_in, const int* in_sizes, int n_in,
                              void* d_out, int out_size, void* d_ws,
                              size_t ws_size, hipStream_t stream) {
  const float* x   = (const float*)d_in[0];
  const int*   dn  = (const int*)d_in[1];
  const float* h0  = (const float*)d_in[2];
  const float* c0  = (const float*)d_in[3];
  const float* Wih = (const float*)d_in[4];
  const float* Whh = (const float*)d_in[5];
  const float* bih = (const float*)d_in[6];
  const float* bhh = (const float*)d_in[7];
  const float* Wpi = (const float*)d_in[8];
  const float* bpi = (const float*)d_in[9];
  const float* Wv  = (const float*)d_in[10];
  const float* bv  = (const float*)d_in[11];
  float* out = (float*)d_out;

  const size_t need = (size_t)T_ * B_ * G4 * sizeof(float);  // 256 MiB
  float* gpre = (float*)d_ws;

  if (ws_size >= need && gpre != nullptr) {
    pre_gemm<<<dim3(8192), dim3(256), 0, stream>>>(x, Wih, bih, bhh, gpre);
    lstm_seq<1><<<dim3(16), dim3(256), 0, stream>>>(
        x, dn, h0, c0, Wih, Whh, bih, bhh, Wpi, bpi, Wv, bv, gpre, out);
  } else {
    lstm_seq<0><<<dim3(16), dim3(256), 0, stream>>>(
        x, dn, h0, c0, Wih, Whh, bih, bhh, Wpi, bpi, Wv, bv, gpre, out);
  }
}